// ModelNew_58798102282515
// MI455X (gfx1250) — compile-verified
//
#include <hip/hip_runtime.h>
#include <hip/hip_bf16.h>

typedef __attribute__((ext_vector_type(2))) float v2f;
typedef __attribute__((ext_vector_type(4))) float v4f;
typedef __attribute__((ext_vector_type(8))) float v8f;

#define B_SZ   64
#define CIN    128
#define COUT   256
#define HW     (192 * 192)          // 36864 elements per (b, ci) row
#define SCALE  (1.0f / (194.0f * 194.0f))

// ---------------------------------------------------------------------------
// Kernel 1: sum_x[b,ci] = sum over H*W of x[b,ci,:,:]
// One 256-thread workgroup per (b,ci) row. 36864 floats = 9216 float4 = 36
// float4 per thread, fully coalesced b128 loads with NT temporal hint
// (single-pass 1.21 GB stream -> keep it out of L2). Four independent
// accumulators break the VALU dependency chain so the loop is load-limited.
// Deterministic LDS tree reduction (bit-stable across replays).
// ---------------------------------------------------------------------------
__global__ void reduce_x_kernel(const float* __restrict__ x,
                                float* __restrict__ sum_x) {
    __shared__ float sdata[256];
    const int row = blockIdx.x;                       // b*CIN + ci, 0..8191
    const v4f* xr = (const v4f*)(x + (size_t)row * HW);

    v4f acc = {0.f, 0.f, 0.f, 0.f};
#pragma unroll
    for (int i = 0; i < 36; ++i) {
        v4f v = __builtin_nontemporal_load(&xr[threadIdx.x + i * 256]);
        acc += v;
    }
    sdata[threadIdx.x] = (acc.x + acc.y) + (acc.z + acc.w);
    __syncthreads();
#pragma unroll
    for (int s = 128; s > 0; s >>= 1) {
        if (threadIdx.x < (unsigned)s) sdata[threadIdx.x] += sdata[threadIdx.x + s];
        __syncthreads();
    }
    if (threadIdx.x == 0) sum_x[row] = sdata[0];
}

// ---------------------------------------------------------------------------
// Kernel 2: wsum[ci,co] = sum over 3x3 of weight[ci,co,:,:]
// weight is [CIN, COUT, 3, 3] row-major -> element ((ci*COUT+co)*9 + k).
// ---------------------------------------------------------------------------
__global__ void reduce_w_kernel(const float* __restrict__ w,
                                float* __restrict__ wsum) {
    const int idx = blockIdx.x * blockDim.x + threadIdx.x;   // ci*COUT + co
    const float* p = w + (size_t)idx * 9;
    float s = 0.0f;
#pragma unroll
    for (int k = 0; k < 9; ++k) s += p[k];
    wsum[idx] = s;
}

// ---------------------------------------------------------------------------
// Kernel 3: g[b,co] = (sum_x[b,:] . wsum[:,co]) * SCALE + conv_bias + bias
// M=64, N=256, K=128 via V_WMMA_F32_16X16X4_F32. One wave per 16x16 tile,
// 4 M-tiles x 16 N-tiles = 64 waves = 8 blocks x 256 threads. 32 chained
// WMMAs per wave over K (steps of 4). EXEC all-ones (full waves).
//
// A (16x4 f32, 2 VGPR/lane): lanes 0-15 -> M=lane, K={k0,k0+1};
//                            lanes 16-31 -> M=lane-16, K={k0+2,k0+3}.
// B (4x16 f32, 2 VGPR/lane): lanes 0-15 -> N=lane, K={k0,k0+1};
//                            lanes 16-31 -> N=lane-16, K={k0+2,k0+3}.
// C/D (16x16 f32, 8 VGPR): VGPR r, lane<16 -> (M=r, N=lane);
//                          lane>=16 -> (M=r+8, N=lane-16).
// ---------------------------------------------------------------------------
__global__ void gemm_wmma_kernel(const float* __restrict__ sum_x,
                                 const float* __restrict__ wsum,
                                 const float* __restrict__ conv_bias,
                                 const float* __restrict__ bias,
                                 float* __restrict__ g) {
    const int wave = (blockIdx.x * blockDim.x + threadIdx.x) >> 5;  // 0..63
    const int lane = threadIdx.x & 31;
    const int mt = wave & 3;          // M tile index (4 tiles of 16 rows)
    const int nt = wave >> 2;         // N tile index (16 tiles of 16 cols)
    const int b0  = mt * 16;
    const int co0 = nt * 16;
    const int half = lane >> 4;       // 0: lanes 0-15, 1: lanes 16-31
    const int lr   = lane & 15;

    v8f c = {};
#pragma unroll
    for (int k0 = 0; k0 < CIN; k0 += 4) {
        const int ka = k0 + 2 * half;
        v2f a, b;
        a.x = sum_x[(b0 + lr) * CIN + ka];
        a.y = sum_x[(b0 + lr) * CIN + ka + 1];
        b.x = wsum[(size_t)(ka    ) * COUT + co0 + lr];
        b.y = wsum[(size_t)(ka + 1) * COUT + co0 + lr];
        // 8 args: (neg_a, A, neg_b, B, c_mod, C, reuse_a, reuse_b)
        c = __builtin_amdgcn_wmma_f32_16x16x4_f32(
                false, a, false, b, (short)0, c, false, false);
    }

    const int co = co0 + lr;
    const float bb = conv_bias[co] + bias[co];
#pragma unroll
    for (int r = 0; r < 8; ++r) {
        const int row = b0 + r + 8 * half;
        g[row * COUT + co] = c[r] * SCALE + bb;
    }
}

// ---------------------------------------------------------------------------
// Kernel 4: out[b] = 10 * logsumexp(g[b, :]) over 256 cols.
// One 256-thread workgroup per row, deterministic LDS reductions.
// ---------------------------------------------------------------------------
__global__ void lse_kernel(const float* __restrict__ g,
                           float* __restrict__ out) {
    __shared__ float sm[256];
    const int b = blockIdx.x;
    const float v = g[b * COUT + threadIdx.x];

    sm[threadIdx.x] = v;
    __syncthreads();
#pragma unroll
    for (int s = 128; s > 0; s >>= 1) {
        if (threadIdx.x < (unsigned)s)
            sm[threadIdx.x] = fmaxf(sm[threadIdx.x], sm[threadIdx.x + s]);
        __syncthreads();
    }
    const float m = sm[0];
    __syncthreads();

    sm[threadIdx.x] = expf(v - m);
    __syncthreads();
#pragma unroll
    for (int s = 128; s > 0; s >>= 1) {
        if (threadIdx.x < (unsigned)s)
            sm[threadIdx.x] += sm[threadIdx.x + s];
        __syncthreads();
    }
    if (threadIdx.x == 0) out[b] = 10.0f * (m + logf(sm[0]));
}

// ---------------------------------------------------------------------------
extern "C" void kernel_launch(void* const* d_in, const int* in_sizes, int n_in,
                              void* d_out, int out_size, void* d_ws, size_t ws_size,
                              hipStream_t stream) {
    const float* x         = (const float*)d_in[0];  // [64,128,192,192]
    const float* weight    = (const float*)d_in[1];  // [128,256,3,3]
    const float* conv_bias = (const float*)d_in[2];  // [256]
    const float* bias      = (const float*)d_in[3];  // [256,1,1] -> flat 256
    float* out = (float*)d_out;                      // [64,1]

    float* sum_x = (float*)d_ws;                     // 64*128   = 8192 f
    float* wsum  = sum_x + B_SZ * CIN;               // 128*256  = 32768 f
    float* g     = wsum + CIN * COUT;                // 64*256   = 16384 f

    reduce_x_kernel<<<B_SZ * CIN, 256, 0, stream>>>(x, sum_x);
    reduce_w_kernel<<<(CIN * COUT) / 256, 256, 0, stream>>>(weight, wsum);
    gemm_wmma_kernel<<<8, 256, 0, stream>>>(sum_x, wsum, conv_bias, bias, g);
    lse_kernel<<<B_SZ, 256, 0, stream>>>(g, out);
}